// ChamferLoss_51127290692351
// MI455X (gfx1250) — compile-verified
//
#include <hip/hip_runtime.h>
#include <math.h>

typedef __attribute__((ext_vector_type(2))) float v2f;
typedef __attribute__((ext_vector_type(8))) float v8f;

#define CH_EPS 1e-6f

// One workgroup (256 threads = 8 waves) per batch of 1024 x / 1024 y points.
__global__ __launch_bounds__(256)
void ChamferLoss_batch_kernel(const float* __restrict__ x,
                              const float* __restrict__ y,
                              float* __restrict__ partial) {
    __shared__ float    xPad[1024 * 4];   // x padded to float4 (x0,x1,x2,0)
    __shared__ float    yPad[1024 * 4];   // y padded to float4
    __shared__ float    x2s[1024];        // ||x_j||^2
    __shared__ float    y2s[1024];        // ||y_i||^2
    __shared__ float    rowMinS[1024];    // min over j of clamped sq-dist (per y row)
    __shared__ unsigned colBits[1024];    // min over i, as uint bits (values >= 0)
    __shared__ float    red[256];

    const int tid = threadIdx.x;
    const int b   = blockIdx.x;
    const float* __restrict__ xg = x + (size_t)b * 3072;  // 1024*3
    const float* __restrict__ yg = y + (size_t)b * 3072;

    // Stage points into LDS (padded), compute squared norms, init col mins.
    for (int p = tid; p < 1024; p += 256) {
        float a0 = xg[3 * p], a1 = xg[3 * p + 1], a2 = xg[3 * p + 2];
        xPad[4 * p] = a0; xPad[4 * p + 1] = a1; xPad[4 * p + 2] = a2; xPad[4 * p + 3] = 0.f;
        x2s[p] = a0 * a0 + a1 * a1 + a2 * a2;
        float c0 = yg[3 * p], c1 = yg[3 * p + 1], c2 = yg[3 * p + 2];
        yPad[4 * p] = c0; yPad[4 * p + 1] = c1; yPad[4 * p + 2] = c2; yPad[4 * p + 3] = 0.f;
        y2s[p] = c0 * c0 + c1 * c1 + c2 * c2;
        colBits[p] = 0x7F800000u;  // +inf bits
    }
    __syncthreads();

    const int lane   = tid & 31;
    const int wave   = tid >> 5;
    const int laneLo = lane & 15;
    const int half   = lane >> 4;       // 0: K=0..1 / M=r ; 1: K=2..3 / M=8+r
    const int kOff   = half << 1;

    // Tile space: 64x64 tiles of 16x16. Each wave owns row-tiles ti = wave + 8k.
    for (int ti = wave; ti < 64; ti += 8) {
        const int aRow = ti * 16 + laneLo;
        // A fragment (y, 16x4 f32): lane holds 2 consecutive K values of its row.
        v2f afrag = *(const v2f*)&yPad[aRow * 4 + kOff];

        float y2r[8];
#pragma unroll
        for (int r = 0; r < 8; ++r) y2r[r] = y2s[ti * 16 + half * 8 + r];

        float rowAcc[8];
#pragma unroll
        for (int r = 0; r < 8; ++r) rowAcc[r] = __builtin_inff();

        for (int tj = 0; tj < 64; ++tj) {
            const int col = tj * 16 + laneLo;
            // B fragment (x, 4x16 f32): same padded-float4 addressing as A.
            v2f bfrag = *(const v2f*)&xPad[col * 4 + kOff];
            v8f c = {};
            // c[M][N] = sum_k y[i][k] * x[j][k]   (K padded 3 -> 4 with zeros)
            c = __builtin_amdgcn_wmma_f32_16x16x4_f32(
                    /*neg_a=*/false, afrag, /*neg_b=*/false, bfrag,
                    /*c_mod=*/(short)0, c, /*reuse_a=*/false, /*reuse_b=*/false);

            const float x2j = x2s[col];
            float cmin = __builtin_inff();
#pragma unroll
            for (int r = 0; r < 8; ++r) {
                float s = fmaf(-2.0f, c[r], y2r[r] + x2j);  // ||y||^2+||x||^2-2yx
                s = fmaxf(s, 0.0f);                          // clamp (commutes w/ min)
                rowAcc[r] = fminf(rowAcc[r], s);
                cmin      = fminf(cmin, s);
            }
            // Column min: values >= 0 so uint-bit atomicMin == float min (exact,
            // order-independent -> deterministic). Lanes L and L+16 share col.
            atomicMin(&colBits[col], __float_as_uint(cmin));
        }

        // Row mins: reduce across the 16 lanes of each half, store (row owned
        // exclusively by this wave -> plain store, no atomic).
#pragma unroll
        for (int r = 0; r < 8; ++r) {
            float v = rowAcc[r];
            v = fminf(v, __shfl_xor(v, 1, 32));
            v = fminf(v, __shfl_xor(v, 2, 32));
            v = fminf(v, __shfl_xor(v, 4, 32));
            v = fminf(v, __shfl_xor(v, 8, 32));
            if (laneLo == 0) rowMinS[ti * 16 + half * 8 + r] = v;
        }
    }
    __syncthreads();

    // sqrt(eps + min) only at the end (monotone), sum rows + cols.
    float local = 0.f;
    for (int i = tid; i < 1024; i += 256) {
        local += sqrtf(CH_EPS + rowMinS[i]);
        local += sqrtf(CH_EPS + __uint_as_float(colBits[i]));
    }
    red[tid] = local;
    __syncthreads();
    if (tid == 0) {
        float s = 0.f;
        for (int k = 0; k < 256; ++k) s += red[k];  // fixed order: deterministic
        partial[b] = s;
    }
}

__global__ void ChamferLoss_finalize_kernel(const float* __restrict__ partial,
                                            float* __restrict__ out,
                                            int nb, float inv) {
    if (threadIdx.x == 0 && blockIdx.x == 0) {
        float s = 0.f;
        for (int k = 0; k < nb; ++k) s += partial[k];  // fixed order
        out[0] = s * inv;  // mean(min1) + mean(min2), each over nb*1024 elems
    }
}

extern "C" void kernel_launch(void* const* d_in, const int* in_sizes, int n_in,
                              void* d_out, int out_size, void* d_ws, size_t ws_size,
                              hipStream_t stream) {
    const float* x = (const float*)d_in[0];
    const float* y = (const float*)d_in[1];
    float* partial = (float*)d_ws;

    const int nb = in_sizes[0] / (1024 * 3);  // B*T = 32 for the reference shapes
    ChamferLoss_batch_kernel<<<nb, 256, 0, stream>>>(x, y, partial);
    ChamferLoss_finalize_kernel<<<1, 1, 0, stream>>>(
        partial, (float*)d_out, nb, 1.0f / (nb * 1024.0f));
}